// PaddingBottleneck_25520695672959
// MI455X (gfx1250) — compile-verified
//
#include <hip/hip_runtime.h>
#include <cstdint>
#include <math.h>

// ---------------------------------------------------------------------------
// PaddingBottleneck for MI455X (gfx1250).
// Memory-bound: 64 MiB read + 64 MiB write, ~0.25 flop/byte -> ~5.8 us at
// 23.3 TB/s. Single HBM pass: stage a 512x32 fp32 column tile (64 KB) into
// LDS with CDNA5 async global->LDS copies (ASYNCcnt / s_wait_asynccnt),
// reduce |x| down the columns in LDS, then rescale in place and stream
// coalesced float4 stores out.
// ---------------------------------------------------------------------------

namespace {
constexpr int kB       = 32;
constexpr int kT       = 1024;              // row stride in floats
constexpr int kD       = 512;               // rows per batch
constexpr int kTT      = 32;                // t-columns per workgroup tile
constexpr int kThreads = 256;               // 8 waves (wave32)
constexpr int kTilesPerBatch = kT / kTT;    // 32
constexpr int kChunks  = kD * kTT * 4 / 16; // 4096 16-byte chunks per tile
}

__global__ __launch_bounds__(kThreads)
void padding_bottleneck_kernel(const float* __restrict__ x,
                               const float* __restrict__ padding_amount,
                               float* __restrict__ out)
{
    // Exactly 64 KB; the only LDS object, so its LDS byte offset is 0.
    __shared__ float tile[kD * kTT];

    const int tid = threadIdx.x;
    const int bid = blockIdx.x;
    const int b   = bid / kTilesPerBatch;
    const int t0  = (bid % kTilesPerBatch) * kTT;
    const size_t gbase = (size_t)b * (size_t)(kD * kT) + (size_t)t0;

    // ---- Phase A: async global -> LDS (tracked by ASYNCcnt) ----------------
    // 4096 chunks of 16 B; 8 chunks per 128-B row segment. GVS addressing:
    // uniform 64-bit base in SGPRs, per-lane 32-bit byte offset in one VGPR.
    // Per k-step: global offset += 32 rows * 4 KB = 128 KB, LDS offset += 4 KB.
    {
        const unsigned long long gs =
            (unsigned long long)(uintptr_t)(x + gbase);
        unsigned voff = ((unsigned)(tid >> 3) * (unsigned)(kT * 4)) +
                        ((unsigned)(tid & 7) * 16u);   // row*4096 + col*16
        unsigned lo   = (unsigned)tid * 16u;           // LDS byte address
        #pragma unroll
        for (int k = 0; k < kChunks / kThreads; ++k) {
            asm volatile("global_load_async_to_lds_b128 %0, %1, %2"
                         :: "v"(lo), "v"(voff), "s"(gs)
                         : "memory");
            voff += (unsigned)((kThreads / 8) * kT * 4);  // 131072 B
            lo   += (unsigned)(kThreads * 16);            // 4096 B
        }
    }
    asm volatile("s_wait_asynccnt 0x0" ::: "memory");
    __syncthreads();

    // ---- Phase B: per-column abs-sum reduction + weights (wave 0) ----------
    // Lane l owns column l: reads tile[d*32 + l] -> 32 consecutive dwords per
    // step, bank-conflict free. v_add_f32 with |src| modifier per element.
    if (tid < 32) {
        const int col = tid;
        float s = 0.0f;
        #pragma unroll 8
        for (int d = 1; d < kD; ++d)
            s += fabsf(tile[d * kTT + col]);
        const float pad = tile[col];
        // 1 - sigmoid(pad - s) == 1 / (1 + exp(pad - s))
        const float w = 1.0f / (1.0f + expf(pad - s));
        const float scale = 1.0f + fabsf(padding_amount[0]);
        out[gbase + col] = pad * scale;      // scaled padding row (coalesced)
        tile[col] = w;                       // reuse row 0 as weight row
    }
    __syncthreads();

    // ---- Phase C: rescale signal rows and store (float4, coalesced) --------
    const float4* tile4 = reinterpret_cast<const float4*>(tile);
    const float4  w4    = tile4[tid & 7];    // c4 == (8+tid+k*256)&7 == tid&7
    for (int i4 = 8 + tid; i4 < (kD * kTT) / 4; i4 += kThreads) {
        const int d  = i4 >> 3;
        const int c4 = i4 & 7;
        const float4 v = tile4[i4];          // ds_load_b128
        float4 r;
        r.x = v.x * w4.x;
        r.y = v.y * w4.y;
        r.z = v.z * w4.z;
        r.w = v.w * w4.w;
        *reinterpret_cast<float4*>(out + gbase + (size_t)d * kT + c4 * 4) = r;
    }
}

extern "C" void kernel_launch(void* const* d_in, const int* in_sizes, int n_in,
                              void* d_out, int out_size, void* d_ws, size_t ws_size,
                              hipStream_t stream) {
    (void)in_sizes; (void)n_in; (void)out_size; (void)d_ws; (void)ws_size;
    const float* x  = (const float*)d_in[0];
    const float* pa = (const float*)d_in[1];
    float* out      = (float*)d_out;

    dim3 grid(kB * kTilesPerBatch);   // 1024 workgroups
    dim3 block(kThreads);             // 8 waves of 32
    padding_bottleneck_kernel<<<grid, block, 0, stream>>>(x, pa, out);
}